// DS_Block_24111946400596
// MI455X (gfx1250) — compile-verified
//
#include <hip/hip_runtime.h>
#include <math.h>

// ---------------------------------------------------------------------------
// Problem constants (match reference)
// ---------------------------------------------------------------------------
#define Bn     16
#define Cn     128
#define Nn     2048
#define KNN    9
#define EPSn   1e-5f
#define BNSC   0.99999500003749969f   // 1/sqrt(1+1e-5)  (eval-mode BN scale)

typedef float v2f __attribute__((ext_vector_type(2)));
typedef float v8f __attribute__((ext_vector_type(8)));

#define WMMA_F32(A, Bf, Cc) \
    __builtin_amdgcn_wmma_f32_16x16x4_f32(false, (A), false, (Bf), (short)0, (Cc), false, false)

// ---------------------------------------------------------------------------
// WMMA GEMM:  Y[b] = epilogue( W[128 x K] @ X[b][K x 2048] + bias )
//   EPI=0 : y = acc + bias (bias may be null)
//   EPI=1 : y = relu( bnG*BNSC*(acc+bias) + bnB )      (fused eval-BN + ReLU)
// One wave -> 32x64 tile (2 M-tiles x 4 N-tiles, 8 accumulators): per K=4 step
// 4 A-loads + 8 B-loads feed 8 v_wmma_f32_16x16x4_f32 (1.5 loads/wmma).
// blockDim=128 (4 waves cover M=128), grid = (1, 2048/64, nbatch)
// ---------------------------------------------------------------------------
template<int EPI>
__global__ __launch_bounds__(128)
void gemm_wmma(const float* __restrict__ W, const float* __restrict__ X,
               const float* __restrict__ bias, const float* __restrict__ bnG,
               const float* __restrict__ bnB, float* __restrict__ Y,
               int K, long xStride, long yStride)
{
    const int lane = threadIdx.x & 31, wave = threadIdx.x >> 5;
    const int lh = lane >> 4, lm = lane & 15;
    const int m0 = wave * 32;                 // 4 waves x 32 rows = M=128
    const int n0 = blockIdx.y * 64;
    const float* Xb = X + (long)blockIdx.z * xStride;
    float* Yb = Y + (long)blockIdx.z * yStride;

    v8f acc[2][4] = {};
    for (int k0 = 0; k0 < K; k0 += 4) {
        const int kk = k0 + lh * 2;           // A/B K-pair per lane-half (ISA 7.12.2)
        const float* wpL = W + (long)(m0 + lm) * K + kk;
        const float* wpH = wpL + 16 * K;
        v2f avL; avL.x = wpL[0]; avL.y = wpL[1];
        v2f avH; avH.x = wpH[0]; avH.y = wpH[1];
        const float* xp = Xb + (long)kk * Nn + n0 + lm;
        v2f b0, b1, b2, b3;
        b0.x = xp[0];  b0.y = xp[Nn + 0];
        b1.x = xp[16]; b1.y = xp[Nn + 16];
        b2.x = xp[32]; b2.y = xp[Nn + 32];
        b3.x = xp[48]; b3.y = xp[Nn + 48];
        acc[0][0] = WMMA_F32(avL, b0, acc[0][0]);
        acc[1][0] = WMMA_F32(avH, b0, acc[1][0]);
        acc[0][1] = WMMA_F32(avL, b1, acc[0][1]);
        acc[1][1] = WMMA_F32(avH, b1, acc[1][1]);
        acc[0][2] = WMMA_F32(avL, b2, acc[0][2]);
        acc[1][2] = WMMA_F32(avH, b2, acc[1][2]);
        acc[0][3] = WMMA_F32(avL, b3, acc[0][3]);
        acc[1][3] = WMMA_F32(avH, b3, acc[1][3]);
    }
#pragma unroll
    for (int p = 0; p < 2; ++p) {
#pragma unroll
        for (int v = 0; v < 8; ++v) {
            const int row = m0 + p * 16 + v + lh * 8;
            const float bi = bias ? bias[row] : 0.f;
            float s = 1.f, t = 0.f;
            if (EPI == 1) { s = bnG[row] * BNSC; t = bnB[row]; }
#pragma unroll
            for (int nt = 0; nt < 4; ++nt) {
                float y = acc[p][nt][v] + bi;
                if (EPI == 1) y = fmaxf(fmaf(s, y, t), 0.f);
                Yb[(long)row * Nn + n0 + nt * 16 + lm] = y;
            }
        }
    }
}

// ---------------------------------------------------------------------------
// Gram matrix (single batch): G[2048x2048] = Xb^T @ Xb, Xb = [128 x 2048]
// A[m][k] = Xb[k][m].  One wave -> 32x64 tile.  grid = (2048/128, 2048/64)
// ---------------------------------------------------------------------------
__global__ __launch_bounds__(128)
void gram_wmma(const float* __restrict__ Xb, float* __restrict__ G)
{
    const int lane = threadIdx.x & 31, wave = threadIdx.x >> 5;
    const int lh = lane >> 4, lm = lane & 15;
    const int m0 = (blockIdx.x * 4 + wave) * 32;
    const int n0 = blockIdx.y * 64;

    v8f acc[2][4] = {};
    for (int k0 = 0; k0 < Cn; k0 += 4) {
        const int kk = k0 + lh * 2;
        const float* ap = Xb + (long)kk * Nn + m0 + lm;
        v2f avL; avL.x = ap[0];  avL.y = ap[Nn + 0];
        v2f avH; avH.x = ap[16]; avH.y = ap[Nn + 16];
        const float* xp = Xb + (long)kk * Nn + n0 + lm;
        v2f b0, b1, b2, b3;
        b0.x = xp[0];  b0.y = xp[Nn + 0];
        b1.x = xp[16]; b1.y = xp[Nn + 16];
        b2.x = xp[32]; b2.y = xp[Nn + 32];
        b3.x = xp[48]; b3.y = xp[Nn + 48];
        acc[0][0] = WMMA_F32(avL, b0, acc[0][0]);
        acc[1][0] = WMMA_F32(avH, b0, acc[1][0]);
        acc[0][1] = WMMA_F32(avL, b1, acc[0][1]);
        acc[1][1] = WMMA_F32(avH, b1, acc[1][1]);
        acc[0][2] = WMMA_F32(avL, b2, acc[0][2]);
        acc[1][2] = WMMA_F32(avH, b2, acc[1][2]);
        acc[0][3] = WMMA_F32(avL, b3, acc[0][3]);
        acc[1][3] = WMMA_F32(avH, b3, acc[1][3]);
    }
#pragma unroll
    for (int p = 0; p < 2; ++p) {
#pragma unroll
        for (int v = 0; v < 8; ++v) {
            const int row = m0 + p * 16 + v + lh * 8;
#pragma unroll
            for (int nt = 0; nt < 4; ++nt)
                G[(long)row * Nn + n0 + nt * 16 + lm] = acc[p][nt][v];
        }
    }
}

// ---------------------------------------------------------------------------
// InstanceNorm (over N) + eval-BN + ReLU (+ optional residual add before ReLU)
// one block per (b,c) row;  MODE 0: relu(bn(in(x)))  MODE 1: relu(bn(in(x))+skip)
// ---------------------------------------------------------------------------
template<int MODE>
__global__ __launch_bounds__(256)
void inorm_bn_kernel(const float* __restrict__ H, const float* __restrict__ skip,
                     const float* __restrict__ g, const float* __restrict__ bb,
                     float* __restrict__ Y)
{
    const int bc = blockIdx.x;
    const int c  = bc & (Cn - 1);
    const float* row = H + (long)bc * Nn;
    __shared__ float s1[256], s2[256];
    float su = 0.f, sq2 = 0.f;
    for (int n = threadIdx.x; n < Nn; n += 256) { float v = row[n]; su += v; sq2 = fmaf(v, v, sq2); }
    s1[threadIdx.x] = su; s2[threadIdx.x] = sq2; __syncthreads();
    for (int st = 128; st > 0; st >>= 1) {
        if (threadIdx.x < st) { s1[threadIdx.x] += s1[threadIdx.x + st]; s2[threadIdx.x] += s2[threadIdx.x + st]; }
        __syncthreads();
    }
    const float mean = s1[0] * (1.f / Nn);
    const float var  = s2[0] * (1.f / Nn) - mean * mean;
    const float inv  = rsqrtf(var + EPSn);
    const float scv  = g[c] * BNSC, shv = bb[c];
    for (int n = threadIdx.x; n < Nn; n += 256) {
        float y = (row[n] - mean) * inv * scv + shv;
        if (MODE == 1) y += skip[(long)bc * Nn + n];
        Y[(long)bc * Nn + n] = fmaxf(y, 0.f);
    }
}

// ---------------------------------------------------------------------------
// Small elementwise / reduction kernels
// ---------------------------------------------------------------------------
__global__ void ew_add_kernel(const float* __restrict__ a, const float* __restrict__ b,
                              float* __restrict__ o)
{
    long i = (long)blockIdx.x * 256 + threadIdx.x;
    o[i] = a[i] + b[i];
}

__global__ void sqnorm_kernel(const float* __restrict__ xl, float* __restrict__ sq)
{
    long i = (long)blockIdx.x * 256 + threadIdx.x;      // i in [0, B*N)
    int b = (int)(i / Nn), n = (int)(i % Nn);
    const float* p = xl + (long)b * Cn * Nn + n;
    float s = 0.f;
    for (int c = 0; c < Cn; ++c) { float v = p[(long)c * Nn]; s = fmaf(v, v, s); }
    sq[i] = s;
}

__global__ __launch_bounds__(256)
void meanpool_kernel(const float* __restrict__ X, float* __restrict__ ga)
{
    const int bc = blockIdx.x;
    const float* row = X + (long)bc * Nn;
    __shared__ float sh[256];
    float s = 0.f;
    for (int n = threadIdx.x; n < Nn; n += 256) s += row[n];
    sh[threadIdx.x] = s; __syncthreads();
    for (int st = 128; st > 0; st >>= 1) {
        if (threadIdx.x < st) sh[threadIdx.x] += sh[threadIdx.x + st];
        __syncthreads();
    }
    if (threadIdx.x == 0) ga[bc] = sh[0] * (1.f / Nn);
}

// ---------------------------------------------------------------------------
// Top-9 per row of score[m] = 2*G[n,m] - sq[m] (row-constant -sq[n] dropped:
// doesn't affect indices).  Tie -> lowest index (matches jax top_k).
// ---------------------------------------------------------------------------
__global__ __launch_bounds__(256)
void topk_kernel(const float* __restrict__ G, const float* __restrict__ sq,
                 int* __restrict__ idxo)
{
    const int n = blockIdx.x;
    __shared__ float sc[Nn];
    __shared__ float rv[256];
    __shared__ int   ri[256];
    const float* g = G + (long)n * Nn;
    for (int m = threadIdx.x; m < Nn; m += 256) sc[m] = 2.f * g[m] - sq[m];
    __syncthreads();
    for (int kk = 0; kk < KNN; ++kk) {
        float bv = -3.0e38f; int bi = 0x7fffffff;
        for (int m = threadIdx.x; m < Nn; m += 256) {
            float v = sc[m];
            if (v > bv) { bv = v; bi = m; }          // strict > keeps lowest m
        }
        rv[threadIdx.x] = bv; ri[threadIdx.x] = bi; __syncthreads();
        for (int st = 128; st > 0; st >>= 1) {
            if (threadIdx.x < st) {
                float ov = rv[threadIdx.x + st]; int oi = ri[threadIdx.x + st];
                if (ov > rv[threadIdx.x] || (ov == rv[threadIdx.x] && oi < ri[threadIdx.x])) {
                    rv[threadIdx.x] = ov; ri[threadIdx.x] = oi;
                }
            }
            __syncthreads();
        }
        if (threadIdx.x == 0) { idxo[n * KNN + kk] = ri[0]; sc[ri[0]] = -3.0e38f; }
        __syncthreads();
    }
}

// ---------------------------------------------------------------------------
// Weight repack:
//   wA[o,c]      = sum_t (d1W[o,c,t] + d1W[o,c+128,t])   (center path of d1)
//   wN[t][o,c]   = d1W[o,c+128,t]                         (neighbor path)
//   w2r[o,t*128+c] = d2W[o,c,t]                           (d2 as one K=384 GEMM)
// ---------------------------------------------------------------------------
__global__ void wprep_kernel(const float* __restrict__ d1W, const float* __restrict__ d2W,
                             float* __restrict__ wA, float* __restrict__ wN,
                             float* __restrict__ w2r)
{
    const int i = blockIdx.x * 256 + threadIdx.x;       // < 128*128
    const int o = i >> 7, c = i & 127;
    const float* wc = d1W + (long)o * 768 + c * 3;
    const float* wn = d1W + (long)o * 768 + (c + 128) * 3;
    float a = 0.f;
#pragma unroll
    for (int t = 0; t < 3; ++t) {
        a += wc[t] + wn[t];
        wN[(long)t * Cn * Cn + i]   = wn[t];
        w2r[(long)o * 384 + t * 128 + c] = d2W[(long)o * 384 + c * 3 + t];
    }
    wA[i] = a;
}

// ---------------------------------------------------------------------------
// d1 gather-combine + BN + ReLU:
//   dh[b][j][o][n] = relu( s*(P[b,o,n] - sum_t Q[t][b,o, idx[b,n,3j+t]]) + sh )
// grid = (N/256, C, B); Q row (8 KB) is block-resident in cache.
// ---------------------------------------------------------------------------
__global__ __launch_bounds__(256)
void d1_combine_kernel(const float* __restrict__ P, const float* __restrict__ Q,
                       const int* __restrict__ idx, const float* __restrict__ g,
                       const float* __restrict__ bb, float* __restrict__ dh)
{
    const int n = blockIdx.x * 256 + threadIdx.x;
    const int o = blockIdx.y, b = blockIdx.z;
    const float base = P[((long)b * Cn + o) * Nn + n];
    const int* id = idx + ((long)b * Nn + n) * KNN;
    const float sc = g[o] * BNSC, sh = bb[o];
#pragma unroll
    for (int j = 0; j < 3; ++j) {
        float acc = base;
#pragma unroll
        for (int t = 0; t < 3; ++t) {
            int m = id[3 * j + t];
            acc -= Q[(((long)t * Bn + b) * Cn + o) * Nn + m];
        }
        dh[(((long)b * 3 + j) * Cn + o) * Nn + n] = fmaxf(fmaf(sc, acc, sh), 0.f);
    }
}

// ---------------------------------------------------------------------------
// Squeeze-excite MLP: ga2[b] = bn2( g2W @ relu(bn1(g1W @ ga[b] + g1b)) + g2b )
// one block per batch, 128 threads.
// ---------------------------------------------------------------------------
__global__ __launch_bounds__(128)
void se_kernel(const float* __restrict__ ga,
               const float* __restrict__ g1W, const float* __restrict__ g1b,
               const float* __restrict__ g1g, const float* __restrict__ g1B,
               const float* __restrict__ g2W, const float* __restrict__ g2b,
               const float* __restrict__ g2g, const float* __restrict__ g2B,
               float* __restrict__ ga2)
{
    const int b = blockIdx.x, t = threadIdx.x;
    __shared__ float vin[128], vmid[32];
    vin[t] = ga[b * 128 + t]; __syncthreads();
    if (t < 32) {
        float acc = g1b[t];
        for (int c = 0; c < 128; ++c) acc = fmaf(g1W[t * 128 + c], vin[c], acc);
        vmid[t] = fmaxf(fmaf(g1g[t] * BNSC, acc, g1B[t]), 0.f);
    }
    __syncthreads();
    float acc = g2b[t];
    for (int c = 0; c < 32; ++c) acc = fmaf(g2W[t * 32 + c], vmid[c], acc);
    ga2[b * 128 + t] = fmaf(g2g[t] * BNSC, acc, g2B[t]);
}

// ---------------------------------------------------------------------------
// Gating: wei = sigmoid(ga2[b,c] + d2out);  out = 2x*wei + 2res*(1-wei)
// ---------------------------------------------------------------------------
__global__ void gate_kernel(const float* __restrict__ x, const float* __restrict__ r,
                            const float* __restrict__ h, const float* __restrict__ ga2,
                            float* __restrict__ out)
{
    long i = (long)blockIdx.x * 256 + threadIdx.x;
    float z = ga2[i / Nn] + h[i];
    float w = 1.f / (1.f + expf(-z));
    out[i] = 2.f * x[i] * w + 2.f * r[i] * (1.f - w);
}

// ---------------------------------------------------------------------------
// Host orchestration
// ---------------------------------------------------------------------------
// d_in flattening assumed (insertion order for top-level dict, then pytree
// alphabetical order inside 'params'):
//   0:x 1:residual
//   be:   2:d1B 3:d1W 4:d1b 5:d1g 6:d2B 7:d2W 8:d2b 9:d2g 10:g1B 11:g1W
//         12:g1b 13:g1g 14:g2B 15:g2W 16:g2b 17:g2g 18:pbB 19:pbW 20:pbb
//         21:pbg 22:pkB 23:pkW 24:pkb 25:pkg
//   res1: 26:bn1b 27:bn1g 28:bn2b 29:bn2g 30:l1W 31:l1b 32:l2W 33:l2b 34:rW 35:rb
//   res2[k] (k=0..2), base = 36+8k: bn1b bn1g bn2b bn2g l1W l1b l2W l2b
extern "C" void kernel_launch(void* const* d_in, const int* in_sizes, int n_in,
                              void* d_out, int out_size, void* d_ws, size_t ws_size,
                              hipStream_t stream)
{
    (void)in_sizes; (void)n_in; (void)out_size; (void)ws_size;
    auto F = [&](int i) { return (const float*)d_in[i]; };
    const float* x   = F(0);
    const float* res = F(1);
    const float *d1B = F(2),  *d1W = F(3),  *d1b = F(4),  *d1g = F(5);
    const float *d2B = F(6),  *d2W = F(7),  *d2b = F(8),  *d2g = F(9);
    const float *g1B = F(10), *g1W = F(11), *g1b = F(12), *g1g = F(13);
    const float *g2B = F(14), *g2W = F(15), *g2b = F(16), *g2g = F(17);
    const float *pbB = F(18), *pbW = F(19), *pbb = F(20), *pbg = F(21);
    const float *pkB = F(22), *pkW = F(23), *pkb = F(24), *pkg = F(25);
    const float *r1bn1b = F(26), *r1bn1g = F(27), *r1bn2b = F(28), *r1bn2g = F(29);
    const float *r1l1W = F(30), *r1l1b = F(31), *r1l2W = F(32), *r1l2b = F(33);
    const float *r1rW  = F(34), *r1rb  = F(35);

    // workspace layout (floats); ~270 MB total
    const long SZ  = (long)Bn * Cn * Nn;   // 4,194,304 (== 2048*2048)
    const long SZb = (long)Cn * Nn;        // per-batch activation slab
    float* ws   = (float*)d_ws;
    float* t0   = ws;                      // x + residual
    float* tmpA = ws + 1 * SZ;             // raw GEMM outputs
    float* tmpB = ws + 2 * SZ;             // post-inorm intermediates
    float* x1   = ws + 3 * SZ;             // resnet1 output
    float* x1b  = ws + 4 * SZ;             // be: after pb conv
    float* xl   = ws + 5 * SZ;             // be: after pk conv
    float* P    = ws + 6 * SZ;             // d1 center path
    float* Q    = ws + 7 * SZ;             // d1 neighbor paths [3][B][C][N]
    float* dh   = ws + 10 * SZ;            // d1 output [B][3][C][N]
    float* d2o  = ws + 13 * SZ;            // d2 output (post bn+relu)
    float* gto  = ws + 14 * SZ;            // gated activations
    float* G    = ws + 15 * SZ;            // per-batch Gram (2048x2048 == SZ)
    float* sq   = ws + 16 * SZ;            // [B*N]
    int*   idx  = (int*)(ws + 16 * SZ + (long)Bn * Nn);          // [B*N*9]
    float* wA   = ws + 16 * SZ + (long)Bn * Nn + (long)Bn * Nn * KNN;
    float* wN   = wA + (long)Cn * Cn;      // 3 * [128x128]
    float* w2r  = wN + 3L * Cn * Cn;       // [128x384]
    float* ga   = w2r + (long)Cn * 384;    // [B*C]
    float* ga2  = ga + (long)Bn * Cn;      // [B*C]

    const dim3 gG(1, Nn / 64, Bn);         // batched 128xKx2048 GEMM (wave = 32x64)
    const dim3 tG(128);
    auto gemm0 = [&](const float* W, const float* X, const float* bias, float* Y,
                     int K, long xs) {
        gemm_wmma<0><<<gG, tG, 0, stream>>>(W, X, bias, nullptr, nullptr, Y, K, xs, SZb);
    };
    auto gemm1 = [&](const float* W, const float* X, const float* bias,
                     const float* g, const float* b2, float* Y, int K, long xs) {
        gemm_wmma<1><<<gG, tG, 0, stream>>>(W, X, bias, g, b2, Y, K, xs, SZb);
    };

    const int ewBlocks = (int)(SZ / 256);

    // ---- resnet1 (pre=True) ----
    ew_add_kernel<<<ewBlocks, 256, 0, stream>>>(x, res, t0);
    gemm0(r1l1W, t0, r1l1b, tmpA, Cn, SZb);
    gemm0(r1rW,  t0, r1rb,  x1,   Cn, SZb);
    inorm_bn_kernel<0><<<Bn * Cn, 256, 0, stream>>>(tmpA, nullptr, r1bn1g, r1bn1b, tmpB);
    gemm0(r1l2W, tmpB, r1l2b, tmpA, Cn, SZb);
    inorm_bn_kernel<1><<<Bn * Cn, 256, 0, stream>>>(tmpA, x1, r1bn2g, r1bn2b, x1);

    // ---- be block front convs (BN fused into GEMM epilogue) ----
    gemm1(pbW, x1,  pbb, pbg, pbB, x1b, Cn, SZb);
    gemm1(pkW, x1b, pkb, pkg, pkB, xl,  Cn, SZb);

    // ---- kNN: per-batch Gram + top-9 (reuses one 16MB G buffer) ----
    sqnorm_kernel<<<(int)(Bn * Nn / 256), 256, 0, stream>>>(xl, sq);
    for (int b = 0; b < Bn; ++b) {
        gram_wmma<<<dim3(Nn / 128, Nn / 64), tG, 0, stream>>>(xl + (long)b * SZb, G);
        topk_kernel<<<Nn, 256, 0, stream>>>(G, sq + (long)b * Nn, idx + (long)b * Nn * KNN);
    }

    // ---- d1 as 4 dense GEMMs + gather-combine; d2 as one K=384 GEMM ----
    wprep_kernel<<<(Cn * Cn) / 256, 256, 0, stream>>>(d1W, d2W, wA, wN, w2r);
    gemm0(wA, xl, d1b, P, Cn, SZb);
    for (int t = 0; t < 3; ++t)
        gemm0(wN + (long)t * Cn * Cn, xl, nullptr, Q + (long)t * SZ, Cn, SZb);
    d1_combine_kernel<<<dim3(Nn / 256, Cn, Bn), 256, 0, stream>>>(P, Q, idx, d1g, d1B, dh);
    gemm1(w2r, dh, d2b, d2g, d2B, d2o, 3 * Cn, 3 * SZb);

    // ---- squeeze-excite + sigmoid gating ----
    meanpool_kernel<<<Bn * Cn, 256, 0, stream>>>(x1b, ga);
    se_kernel<<<Bn, 128, 0, stream>>>(ga, g1W, g1b, g1g, g1B, g2W, g2b, g2g, g2B, ga2);
    gate_kernel<<<ewBlocks, 256, 0, stream>>>(x, res, d2o, ga2, gto);

    // ---- res2 x3 (pre=False); final block writes d_out ----
    const float* cur = gto;
    float* outs[3] = { t0, x1b, (float*)d_out };
    for (int k = 0; k < 3; ++k) {
        const int base = 36 + 8 * k;
        const float *bn1b = F(base + 0), *bn1g = F(base + 1);
        const float *bn2b = F(base + 2), *bn2g = F(base + 3);
        const float *l1W  = F(base + 4), *l1b  = F(base + 5);
        const float *l2W  = F(base + 6), *l2b  = F(base + 7);
        gemm0(l1W, cur, l1b, tmpA, Cn, SZb);
        inorm_bn_kernel<0><<<Bn * Cn, 256, 0, stream>>>(tmpA, nullptr, bn1g, bn1b, tmpB);
        gemm0(l2W, tmpB, l2b, tmpA, Cn, SZb);
        inorm_bn_kernel<1><<<Bn * Cn, 256, 0, stream>>>(tmpA, cur, bn2g, bn2b, outs[k]);
        cur = outs[k];
    }
}